// MaxCutScoreNet_40364102648179
// MI455X (gfx1250) — compile-verified
//
#include <hip/hip_runtime.h>
#include <hip/hip_bf16.h>
#include <math.h>

typedef __attribute__((ext_vector_type(2))) float v2f;
typedef __attribute__((ext_vector_type(8))) float v8f;

static constexpr int   kN     = 50000;
static constexpr int   kE     = 1600000;
static constexpr int   kFIN   = 128;
static constexpr float kDelta = 2.0f;

// ---------------- elementwise / edge kernels ----------------

__global__ void mc_zero_f32(float* __restrict__ p, int n) {
  int i = blockIdx.x * blockDim.x + threadIdx.x;
  if (i < n) p[i] = 0.0f;
}

__global__ void mc_deg(const int* __restrict__ dst, const float* __restrict__ ew,
                       float* __restrict__ deg, int e_count) {
  int e = blockIdx.x * blockDim.x + threadIdx.x;
  if (e < e_count) atomicAdd(&deg[dst[e]], ew[e]);
}

__global__ void mc_dinv(const float* __restrict__ deg, float* __restrict__ dinv, int n) {
  int i = blockIdx.x * blockDim.x + threadIdx.x;
  if (i < n) {
    float d = deg[i];
    dinv[i] = d > 0.0f ? rsqrtf(fmaxf(d, 1e-12f)) : 0.0f;
  }
}

__global__ void mc_edgew(const int* __restrict__ src, const int* __restrict__ dst,
                         const float* __restrict__ ew, const float* __restrict__ dinv,
                         float* __restrict__ w, int e_count) {
  int e = blockIdx.x * blockDim.x + threadIdx.x;
  if (e < e_count) w[e] = kDelta * dinv[src[e]] * ew[e] * dinv[dst[e]];
}

// U lanes cooperate on one edge -> fully coalesced gather + coalesced f32 atomics.
// 32-bit index math only (max index = 50000*32 < 2^31).
template <int U, int LOGU>
__global__ __launch_bounds__(256) void mc_scatter(
    const int* __restrict__ src, const int* __restrict__ dst, const float* __restrict__ w,
    const float* __restrict__ hw, float* __restrict__ agg, int e_count) {
  int t = blockIdx.x * blockDim.x + threadIdx.x;
  int e = t >> LOGU;
  int f = t & (U - 1);
  if (e < e_count) {
    int si = (src[e] << LOGU) + f;
    int di = (dst[e] << LOGU) + f;
    atomicAdd(&agg[di], hw[si] * w[e]);
  }
}

__global__ void mc_tanh(const float* __restrict__ a, float* __restrict__ o, int n) {
  int i = blockIdx.x * blockDim.x + threadIdx.x;
  if (i < n) o[i] = tanhf(a[i]);
}

__global__ void mc_final(const float* __restrict__ h, const float* __restrict__ Wf,
                         const float* __restrict__ bf, float* __restrict__ out, int n) {
  int i = blockIdx.x * blockDim.x + threadIdx.x;
  if (i < n) {
    float acc = bf[0];
#pragma unroll
    for (int k = 0; k < 16; ++k) acc += h[i * 16 + k] * Wf[k];
    out[i] = tanhf(acc);
  }
}

// ---------------- fp32 WMMA GEMM ----------------
// One wave computes MT=4 vertically-stacked 16x16 output tiles sharing one B fragment.
// Per K-step: clause of 4 A-frag loads + 2 B loads -> one wait -> 4 back-to-back
// v_wmma_f32_16x16x4_f32 (independent accumulators, shared B).
// Fragment layouts (ISA 7.12.2):
//   A 16x4 : lane<16 -> M=lane, K=k..k+1 ; lane>=16 -> M=lane-16, K=k+2..k+3
//   B 4x16 : vgpr j, lanes0-15 row K=k+j (N=lane); lanes16-31 row K=k+2+j (N=lane-16)
//   C/D    : vgpr r -> M=r (lanes0-15) / r+8 (lanes16-31), N=lane&15
// FULL = (NF % 16 == 0): no column guard. Otherwise clamp column + 0/1 mask
// (branch-free, loads in-bounds, no exec save/restore).
// EPI: 0 = D+bias ; 1 = conv (hw = D, agg = self_w*D + bias) ; 2 = relu(D+bias)
template <int EPI, bool FULL, int MT>
__global__ __launch_bounds__(256) void mc_gemm_wmma(
    const float* __restrict__ A, const float* __restrict__ W, const float* __restrict__ bias,
    float* __restrict__ Dout, float* __restrict__ Agg,
    int M, int K, int NF, float self_w) {
  const int wave = threadIdx.x >> 5;
  const int lane = threadIdx.x & 31;
  const int half = lane >> 4;
  const int lm   = lane & 15;
  const int ntn  = (NF + 15) >> 4;
  const int mtn  = (M + 15) >> 4;
  const int mchunks = (mtn + MT - 1) / MT;
  const int tile  = blockIdx.x * 8 + wave;
  const int chunk = tile / ntn;
  if (chunk >= mchunks) return;  // wave-uniform exit: EXEC all-1s for WMMA
  const int ntile = tile - chunk * ntn;
  const int n0  = ntile << 4;
  const int col = n0 + lm;

  const int   colc = FULL ? col : (col < NF ? col : NF - 1);
  const float mskf = FULL ? 1.0f : (col < NF ? 1.0f : 0.0f);

  const int mt0 = chunk * MT;
  const float* ap[MT];
#pragma unroll
  for (int t = 0; t < MT; ++t) {
    int mt = mt0 + t;
    if (mt > mtn - 1) mt = mtn - 1;  // clamp: duplicate compute, store-guarded
    ap[t] = A + (size_t)((mt << 4) + lm) * K + half * 2;
  }
  const float* bp = W + (size_t)(half * 2) * NF + colc;
  const size_t bstep = (size_t)4 * NF;

  v8f c[MT];
#pragma unroll
  for (int t = 0; t < MT; ++t) c[t] = (v8f){};

  for (int k = 0; k < K; k += 4) {
    // gather phase: independent destination registers so loads can clause
    v2f b;
    b.x = bp[0];
    b.y = bp[NF];
    v2f a[MT];
#pragma unroll
    for (int t = 0; t < MT; ++t) {
      a[t].x = ap[t][0];
      a[t].y = ap[t][1];
      ap[t] += 4;
    }
    if (!FULL) { b.x *= mskf; b.y *= mskf; }
    bp += bstep;
    // compute phase: 4 back-to-back WMMAs on shared B
#pragma unroll
    for (int t = 0; t < MT; ++t) {
      c[t] = __builtin_amdgcn_wmma_f32_16x16x4_f32(false, a[t], false, b, (short)0, c[t],
                                                   false, false);
    }
  }

  if (!FULL && col >= NF) return;
  const float bv = bias[col];
#pragma unroll
  for (int t = 0; t < MT; ++t) {
    if (mt0 + t >= mtn) break;  // skip clamped duplicates (no double store)
    const int m0 = (mt0 + t) << 4;
#pragma unroll
    for (int r = 0; r < 8; ++r) {
      int row = m0 + r + half * 8;
      size_t idx = (size_t)row * NF + col;
      float v = c[t][r];
      if (EPI == 0) {
        Dout[idx] = v + bv;
      } else if (EPI == 2) {
        float x = v + bv;
        Dout[idx] = x > 0.0f ? x : 0.0f;
      } else {
        Dout[idx] = v;                // message features (no bias)
        Agg[idx]  = self_w * v + bv;  // self-loop + bias folded into accumulator init
      }
    }
  }
}

static inline void launch_gemm(int epi, const float* A, const float* W, const float* bias,
                               float* Dout, float* Agg, int M, int K, int NF, float self_w,
                               hipStream_t s) {
  constexpr int MT = 4;
  int ntn = (NF + 15) / 16;
  int mtn = (M + 15) / 16;
  int mchunks = (mtn + MT - 1) / MT;
  int tiles = mchunks * ntn;
  int blocks = (tiles + 7) / 8;  // 8 waves per 256-thread block
  bool full = (NF % 16) == 0;
  if (full) {
    switch (epi) {
      case 0:  mc_gemm_wmma<0, true, MT><<<blocks, 256, 0, s>>>(A, W, bias, Dout, Agg, M, K, NF, self_w); break;
      case 1:  mc_gemm_wmma<1, true, MT><<<blocks, 256, 0, s>>>(A, W, bias, Dout, Agg, M, K, NF, self_w); break;
      default: mc_gemm_wmma<2, true, MT><<<blocks, 256, 0, s>>>(A, W, bias, Dout, Agg, M, K, NF, self_w); break;
    }
  } else {
    switch (epi) {
      case 0:  mc_gemm_wmma<0, false, MT><<<blocks, 256, 0, s>>>(A, W, bias, Dout, Agg, M, K, NF, self_w); break;
      case 1:  mc_gemm_wmma<1, false, MT><<<blocks, 256, 0, s>>>(A, W, bias, Dout, Agg, M, K, NF, self_w); break;
      default: mc_gemm_wmma<2, false, MT><<<blocks, 256, 0, s>>>(A, W, bias, Dout, Agg, M, K, NF, self_w); break;
    }
  }
}

extern "C" void kernel_launch(void* const* d_in, const int* in_sizes, int n_in,
                              void* d_out, int out_size, void* d_ws, size_t ws_size,
                              hipStream_t stream) {
  (void)in_sizes; (void)n_in; (void)out_size; (void)ws_size;
  const float* x  = (const float*)d_in[0];
  const int* edge_index = (const int*)d_in[1];
  const int* src = edge_index;        // row 0
  const int* dst = edge_index + kE;   // row 1
  const float* ew = (const float*)d_in[2];
  const float* Wi = (const float*)d_in[3];
  const float* bi = (const float*)d_in[4];
  const float* Wc[12];
  const float* bc[12];
  for (int i = 0; i < 12; ++i) {
    Wc[i] = (const float*)d_in[5 + 2 * i];
    bc[i] = (const float*)d_in[6 + 2 * i];
  }
  const float* Wm0 = (const float*)d_in[29];
  const float* bm0 = (const float*)d_in[30];
  const float* Wm1 = (const float*)d_in[31];
  const float* bm1 = (const float*)d_in[32];
  const float* Wf  = (const float*)d_in[33];
  const float* bf  = (const float*)d_in[34];
  float* out = (float*)d_out;

  // carve scratch from d_ws (~71 MB total)
  float* ws = (float*)d_ws;
  size_t off = 0;
  auto carve = [&](size_t elems) {
    float* p = ws + off;
    off += (elems + 63) & ~(size_t)63;
    return p;
  };
  float* deg  = carve(kN);
  float* dinv = carve(kN);
  float* w    = carve(kE);
  float* hA   = carve((size_t)kN * kFIN);
  float* hB   = carve((size_t)kN * kFIN);
  float* hw   = carve((size_t)kN * 32);
  float* agg  = carve((size_t)kN * 32);

  const int TB = 256;
  // normalization weights: deg -> dinv -> per-edge w (deterministic each call)
  mc_zero_f32<<<(kN + TB - 1) / TB, TB, 0, stream>>>(deg, kN);
  mc_deg<<<(kE + TB - 1) / TB, TB, 0, stream>>>(dst, ew, deg, kE);
  mc_dinv<<<(kN + TB - 1) / TB, TB, 0, stream>>>(deg, dinv, kN);
  mc_edgew<<<(kE + TB - 1) / TB, TB, 0, stream>>>(src, dst, ew, dinv, w, kE);

  // h = x @ Wi + bi  (50000 x 128 x 128, fp32 WMMA)
  launch_gemm(0, x, Wi, bi, hA, nullptr, kN, kFIN, kFIN, 0.0f, stream);

  static const int mpu[12] = {32, 32, 32, 32, 16, 16, 16, 16, 8, 8, 8, 8};
  const float self_w = 1.0f - kDelta;
  float* hcur = hA;
  float* hnext = hB;
  int din = kFIN;
  for (int i = 0; i < 12; ++i) {
    int u = mpu[i];
    // hw = h @ Wc ; agg = (1-delta)*hw + bc  (agg fully initialized -> no stale state)
    launch_gemm(1, hcur, Wc[i], bc[i], hw, agg, kN, din, u, self_w, stream);
    long long tot = (long long)kE * u;
    int blocks = (int)((tot + TB - 1) / TB);
    if (u == 32)      mc_scatter<32, 5><<<blocks, TB, 0, stream>>>(src, dst, w, hw, agg, kE);
    else if (u == 16) mc_scatter<16, 4><<<blocks, TB, 0, stream>>>(src, dst, w, hw, agg, kE);
    else              mc_scatter<8, 3><<<blocks, TB, 0, stream>>>(src, dst, w, hw, agg, kE);
    int nel = kN * u;
    mc_tanh<<<(nel + TB - 1) / TB, TB, 0, stream>>>(agg, hnext, nel);
    float* t = hcur; hcur = hnext; hnext = t;
    din = u;
  }

  // MLP: relu(h@Wm0+b) 8->16 ; relu(h@Wm1+b) 16->16
  launch_gemm(2, hcur, Wm0, bm0, hnext, nullptr, kN, 8, 16, 0.0f, stream);
  { float* t = hcur; hcur = hnext; hnext = t; }
  launch_gemm(2, hcur, Wm1, bm1, hnext, nullptr, kN, 16, 16, 0.0f, stream);
  { float* t = hcur; hcur = hnext; hnext = t; }

  // out = tanh(h @ Wf + bf), 16 -> 1
  mc_final<<<(kN + TB - 1) / TB, TB, 0, stream>>>(hcur, Wf, bf, out, kN);
}